// GraphormerLayer_18837726560772
// MI455X (gfx1250) — compile-verified
//
#include <hip/hip_runtime.h>
#include <hip/hip_bf16.h>

// ---------------- problem constants (from reference) ----------------
#define N_NODES 8192
#define DIMF    256
#define N_GRAPHS 16
#define MAXN    512            // nodes per graph == MAX_GRAPH_SIZE
#define N_EDGES 131072
#define HEADS   8
#define DH      32

typedef __attribute__((ext_vector_type(2))) float v2f;
typedef __attribute__((ext_vector_type(8))) float v8f;

// V_WMMA_F32_16X16X4_F32 : D(16x16 f32) = A(16x4 f32) * B(4x16 f32) + C
__device__ __forceinline__ v8f wmma_f32(v2f a, v2f b, v8f c) {
    return __builtin_amdgcn_wmma_f32_16x16x4_f32(false, a, false, b,
                                                 (short)0, c, false, false);
}

// async global -> LDS, 128-bit per lane (GLOBAL_LOAD_ASYNC_TO_LDS_B128,
// tracked by ASYNCcnt). VDST VGPR carries the LDS byte address (= low 32 bits
// of the generic pointer per the ISA aperture rule), VADDR the 64-bit global
// address. Inline asm keeps the src/dst roles explicit and compiler-checkable.
__device__ __forceinline__ void async_g2l_b128(void* lds, const void* g) {
    asm volatile("global_load_async_to_lds_b128 %0, %1, off"
                 :: "v"((unsigned)(unsigned long long)lds),
                    "v"((unsigned long long)g)
                 : "memory");
}

__device__ __forceinline__ void wait_async0() {
#if __has_builtin(__builtin_amdgcn_s_wait_asynccnt)
    __builtin_amdgcn_s_wait_asynccnt(0);
#else
    asm volatile("s_wait_asynccnt 0" ::: "memory");
#endif
}

// ---------------- stage 1: x = node_feature; deg = 0 ----------------
__global__ void k_init(const float* __restrict__ node,
                       float* __restrict__ x, int* __restrict__ deg) {
    int i = blockIdx.x * blockDim.x + threadIdx.x;
    if (i < N_NODES * DIMF) x[i] = node[i];
    if (i < N_NODES) deg[i] = 0;
}

// degree of each source node
__global__ void k_deg(const int* __restrict__ eidx, int* __restrict__ deg) {
    int e = blockIdx.x * blockDim.x + threadIdx.x;
    if (e < N_EDGES) atomicAdd(&deg[eidx[e]], 1);
}

// scatter-add edge features onto source nodes (float4 per thread)
__global__ void k_edge(const float* __restrict__ ef, const int* __restrict__ eidx,
                       float* __restrict__ x) {
    long idx = (long)blockIdx.x * blockDim.x + threadIdx.x;   // E*64 threads
    int e = (int)(idx >> 6);
    int c = ((int)idx & 63) << 2;
    if (e >= N_EDGES) return;
    int s = eidx[e];
    const float4 v = *(const float4*)(ef + (long)e * DIMF + c);
    float* dst = x + (long)s * DIMF + c;
    atomicAdd(dst + 0, v.x);
    atomicAdd(dst + 1, v.y);
    atomicAdd(dst + 2, v.z);
    atomicAdd(dst + 3, v.w);
}

// x += centrality[min(deg, 511)]
__global__ void k_cent(const float* __restrict__ cent, const int* __restrict__ deg,
                       float* __restrict__ x) {
    int i = blockIdx.x * blockDim.x + threadIdx.x;
    if (i >= N_NODES * DIMF) return;
    int n = i >> 8, c = i & 255;
    int d = deg[n]; d = d > (MAXN - 1) ? (MAXN - 1) : d;
    x[i] += cent[d * DIMF + c];
}

// ---------------- generic fp32 WMMA GEMM: C = A[M x Kd] * B + bias ----------
// block tile 64x64, 8 waves x two 16x16 C tiles; double-buffered async staging.
__global__ void __launch_bounds__(256)
k_gemm(const float* __restrict__ A, const float* __restrict__ B,
       const float* __restrict__ bias, float* __restrict__ C,
       int Kd, int ldb, int bcol, int ldc, int zeroFrom) {
    __shared__ float As[2][64][40];   // 40-float stride: 160B rows, b128-aligned
    __shared__ float Bs[2][32][72];   // 72-float stride: 288B rows, b128-aligned
    int t = threadIdx.x;
    int w = t >> 5, lane = t & 31, half = lane >> 4, l16 = lane & 15;
    int m0 = (w & 3) * 16;
    int n0 = (w >> 2) * 32;
    long rowBase = (long)blockIdx.x * 64;
    int colBase = blockIdx.y * 64;
    v8f acc0 = {0, 0, 0, 0, 0, 0, 0, 0};
    v8f acc1 = {0, 0, 0, 0, 0, 0, 0, 0};

    // issue one k-tile: A 64x32 + B 32x64, 2+2 b128 transfers per thread
    auto issue = [&](int bi, int k0) {
#pragma unroll
        for (int i = 0; i < 2; ++i) {
            int idx = t + i * 256;
            int r = idx >> 3, c = (idx & 7) << 2;
            async_g2l_b128(&As[bi][r][c], A + (rowBase + r) * (long)Kd + k0 + c);
        }
#pragma unroll
        for (int i = 0; i < 2; ++i) {
            int idx = t + i * 256;
            int r = idx >> 4, c = (idx & 15) << 2;
            async_g2l_b128(&Bs[bi][r][c],
                           B + (long)(k0 + r) * ldb + bcol + colBase + c);
        }
    };

    int buf = 0;
    issue(buf, 0);
    for (int k0 = 0; k0 < Kd; k0 += 32) {
        wait_async0();            // own async copies done (incl. this tile)
        __syncthreads();          // everyone's copies done; prev reads retired
        if (k0 + 32 < Kd) issue(buf ^ 1, k0 + 32);   // overlap with compute
#pragma unroll
        for (int kk = 0; kk < 32; kk += 4) {
            v2f a, b0, b1;
            a[0] = As[buf][m0 + l16][kk + 2 * half];       // A: K = v + 2*half
            a[1] = As[buf][m0 + l16][kk + 2 * half + 1];
            b0[0] = Bs[buf][kk + 2 * half][n0 + l16];      // B: K = v + 2*half
            b0[1] = Bs[buf][kk + 2 * half + 1][n0 + l16];
            b1[0] = Bs[buf][kk + 2 * half][n0 + 16 + l16];
            b1[1] = Bs[buf][kk + 2 * half + 1][n0 + 16 + l16];
            acc0 = wmma_f32(a, b0, acc0);
            acc1 = wmma_f32(a, b1, acc1);
        }
        buf ^= 1;
    }
#pragma unroll
    for (int v = 0; v < 8; ++v) {               // C: row = v + 8*half, col = l16
        long r = rowBase + m0 + v + 8 * half;
        int c0 = colBase + n0 + l16;
        float x0 = acc0[v], x1 = acc1[v];
        if (bias) { x0 += bias[c0]; x1 += bias[c0 + 16]; }
        if (r >= zeroFrom) { x0 = 0.f; x1 = 0.f; }  // zero last graph (fused)
        C[r * ldc + c0] = x0;
        C[r * ldc + c0 + 16] = x1;
    }
}

// ---------------- attention: one block per (graph b, head h) ----------------
// flash-style online softmax; double-buffered async K/V staging.
__global__ void __launch_bounds__(256)
k_attn(const float* __restrict__ qb, const float* __restrict__ kb,
       const float* __restrict__ vb, const int* __restrict__ dist,
       const float* __restrict__ dist_bias, float* __restrict__ ao) {
    int h = blockIdx.x, b = blockIdx.y;
    __shared__ float Ks[2][64][40];
    __shared__ float Vs[2][64][40];
    __shared__ float Ps[8][16][17];     // per-wave C->A layout bounce buffer
    __shared__ float db[MAXN];          // dist_bias[:, h]
    int t = threadIdx.x;
    int w = t >> 5, lane = t & 31, half = lane >> 4, l16 = lane & 15;
    for (int i = t; i < MAXN; i += 256) db[i] = dist_bias[i * HEADS + h];
    const float scale = 0.1767766952966369f;   // 32^-0.5

    // stage one 64-key supertile of K and V (2+2 b128 transfers per thread)
    auto issue_kv = [&](int bi, int kt) {
#pragma unroll
        for (int i = 0; i < 2; ++i) {
            int idx = t + i * 256;
            int r = idx >> 3, c = (idx & 7) << 2;
            long off = (long)(b * MAXN + kt * 64 + r) * DIMF + h * DH + c;
            async_g2l_b128(&Ks[bi][r][c], kb + off);
            async_g2l_b128(&Vs[bi][r][c], vb + off);
        }
    };

    int buf = 0;
    for (int rnd = 0; rnd < 4; ++rnd) {
        int q0 = rnd * 128 + w * 16;           // this wave's 16 query rows
        // Q as 8 A-fragments (K = dh, steps of 4), kept in registers
        v2f qf[8];
#pragma unroll
        for (int kk = 0; kk < 8; ++kk) {
            const float* qp = qb + (long)(b * MAXN + q0 + l16) * DIMF
                              + h * DH + kk * 4 + 2 * half;
            qf[kk][0] = qp[0];
            qf[kk][1] = qp[1];
        }
        float mrow[8], lrow[8];
        v8f o0 = {0, 0, 0, 0, 0, 0, 0, 0};
        v8f o1 = {0, 0, 0, 0, 0, 0, 0, 0};
#pragma unroll
        for (int v = 0; v < 8; ++v) { mrow[v] = -3.0e38f; lrow[v] = 0.f; }

        issue_kv(buf, 0);
        for (int kt = 0; kt < 8; ++kt) {       // 64-key supertiles
            wait_async0();
            __syncthreads();
            if (kt < 7) issue_kv(buf ^ 1, kt + 1);    // overlap with compute
            for (int sub = 0; sub < 4; ++sub) {
                int key0 = kt * 64 + sub * 16;
                // S = Q K^T  (B fragment: B[d][j] = K[j][d])
                v8f s = {0, 0, 0, 0, 0, 0, 0, 0};
#pragma unroll
                for (int kk = 0; kk < 8; ++kk) {
                    v2f bf;
                    bf[0] = Ks[buf][sub * 16 + l16][kk * 4 + 2 * half];
                    bf[1] = Ks[buf][sub * 16 + l16][kk * 4 + 2 * half + 1];
                    s = wmma_f32(qf[kk], bf, s);
                }
                // scale + distance bias (gather), then online softmax
                float p[8], tm[8];
#pragma unroll
                for (int v = 0; v < 8; ++v) {
                    int row = q0 + v + 8 * half;
                    int col = key0 + l16;
                    int dd = dist[(long)(b * MAXN + row) * MAXN + col];
                    dd = dd > (MAXN - 1) ? (MAXN - 1) : dd;
                    p[v] = s[v] * scale + db[dd];
                }
#pragma unroll
                for (int v = 0; v < 8; ++v) {  // row max across 16-lane half
                    float mv = p[v];
                    mv = fmaxf(mv, __shfl_xor(mv, 1, 32));
                    mv = fmaxf(mv, __shfl_xor(mv, 2, 32));
                    mv = fmaxf(mv, __shfl_xor(mv, 4, 32));
                    mv = fmaxf(mv, __shfl_xor(mv, 8, 32));
                    tm[v] = mv;
                }
#pragma unroll
                for (int v = 0; v < 8; ++v) {
                    float mnew = fmaxf(mrow[v], tm[v]);
                    float alpha = __expf(mrow[v] - mnew);
                    mrow[v] = mnew;
                    float pv = __expf(p[v] - mnew);
                    float rs = pv;
                    rs += __shfl_xor(rs, 1, 32);
                    rs += __shfl_xor(rs, 2, 32);
                    rs += __shfl_xor(rs, 4, 32);
                    rs += __shfl_xor(rs, 8, 32);
                    lrow[v] = lrow[v] * alpha + rs;
                    o0[v] *= alpha;
                    o1[v] *= alpha;
                    Ps[w][v + 8 * half][l16] = pv;   // C layout -> LDS
                }
                // intra-wave LDS RAW: explicit CDNA5 split-counter wait
                asm volatile("s_wait_dscnt 0" ::: "memory");
                // O += P V  (P reloaded in A layout from LDS)
#pragma unroll
                for (int kk = 0; kk < 4; ++kk) {
                    v2f af, bf0, bf1;
                    af[0] = Ps[w][l16][kk * 4 + 2 * half];
                    af[1] = Ps[w][l16][kk * 4 + 2 * half + 1];
                    bf0[0] = Vs[buf][sub * 16 + kk * 4 + 2 * half][l16];
                    bf0[1] = Vs[buf][sub * 16 + kk * 4 + 2 * half + 1][l16];
                    bf1[0] = Vs[buf][sub * 16 + kk * 4 + 2 * half][16 + l16];
                    bf1[1] = Vs[buf][sub * 16 + kk * 4 + 2 * half + 1][16 + l16];
                    o0 = wmma_f32(af, bf0, o0);
                    o1 = wmma_f32(af, bf1, o1);
                }
            }
            buf ^= 1;
        }
        // normalize and write out: ao[node][h*32 + d]
#pragma unroll
        for (int v = 0; v < 8; ++v) {
            float inv = 1.0f / lrow[v];
            int node = b * MAXN + q0 + v + 8 * half;
            ao[(long)node * DIMF + h * DH + l16]      = o0[v] * inv;
            ao[(long)node * DIMF + h * DH + 16 + l16] = o1[v] * inv;
        }
    }
}

// ---------------- host-side orchestration ----------------
extern "C" void kernel_launch(void* const* d_in, const int* in_sizes, int n_in,
                              void* d_out, int out_size, void* d_ws, size_t ws_size,
                              hipStream_t stream) {
    const float* node      = (const float*)d_in[0];
    const float* ef        = (const float*)d_in[1];
    const float* cent      = (const float*)d_in[2];
    const float* dist_bias = (const float*)d_in[3];
    const float* Wq        = (const float*)d_in[4];
    const float* Wkv       = (const float*)d_in[5];
    const float* Wo        = (const float*)d_in[6];
    const float* bo        = (const float*)d_in[7];
    const int*   eidx      = (const int*)d_in[8];   // [2, E]; row 0 = src
    const int*   dist      = (const int*)d_in[10];  // [G, 512, 512]
    (void)in_sizes; (void)n_in; (void)out_size; (void)ws_size;

    const size_t ND = (size_t)N_NODES * DIMF;
    float* x    = (float*)d_ws;
    float* qbuf = x + ND;
    float* kbuf = qbuf + ND;
    float* vbuf = kbuf + ND;
    float* ao   = vbuf + ND;
    int*   deg  = (int*)(ao + ND);

    // x = node + centrality[deg] + scatter(edge_feature)
    k_init<<<(N_NODES * DIMF + 255) / 256, 256, 0, stream>>>(node, x, deg);
    k_deg<<<(N_EDGES + 255) / 256, 256, 0, stream>>>(eidx, deg);
    k_edge<<<(N_EDGES * 64) / 256, 256, 0, stream>>>(ef, eidx, x);
    k_cent<<<(N_NODES * DIMF + 255) / 256, 256, 0, stream>>>(cent, deg, x);

    // Q = x Wq ; K = x Wkv[:, :256] ; V = x Wkv[:, 256:]
    dim3 gg(N_NODES / 64, DIMF / 64);
    k_gemm<<<gg, 256, 0, stream>>>(x, Wq,  nullptr, qbuf, 256, 256, 0,   256, 1 << 30);
    k_gemm<<<gg, 256, 0, stream>>>(x, Wkv, nullptr, kbuf, 256, 512, 0,   256, 1 << 30);
    k_gemm<<<gg, 256, 0, stream>>>(x, Wkv, nullptr, vbuf, 256, 512, 256, 256, 1 << 30);

    // per-(graph, head) attention with distance bias
    k_attn<<<dim3(HEADS, N_GRAPHS), 256, 0, stream>>>(qbuf, kbuf, vbuf, dist,
                                                      dist_bias, ao);

    // pred = ao Wo + bo, with last graph's rows zeroed (reference off-by-one)
    k_gemm<<<gg, 256, 0, stream>>>(ao, Wo, bo, (float*)d_out,
                                   256, 256, 0, 256, N_NODES - MAXN);
}